// EACB_51032801411594
// MI455X (gfx1250) — compile-verified
//
#include <hip/hip_runtime.h>

// CDNA5 / gfx1250 fused 5-branch conv == single 3x3 conv via branch
// re-parameterization, computed as implicit GEMM on V_WMMA_F32_16X16X4_F32.
// B operands are channel-pair-interleaved in LDS (one ds_load_b64 each) and
// software-pipelined 8 deep; input tiles are double-buffered in LDS via
// GLOBAL_LOAD_ASYNC_TO_LDS_B32 so chunk c+1's DMA overlaps chunk c's WMMAs.

typedef __attribute__((ext_vector_type(2))) float v2f;
typedef __attribute__((ext_vector_type(8))) float v8f;

#define C_IN   128
#define C_OUT  128
#define BATCH  16
#define HH     192
#define WW     192
#define TILE   16
#define ICCH   32                  // input channels staged per LDS pass
#define NPAIR  (ICCH / 2)          // 16 channel-pair planes
#define ROW_F2 24                  // padded row stride in float2 (48 dwords)
#define PLANE_F2 (18 * ROW_F2)     // 432 float2 = 864 dwords == 32 (mod 64)
#define BUF_F2 (NPAIR * PLANE_F2)  // 6912 float2 = 55296 B per buffer

// ---------------------------------------------------------------------------
// Phase 1: fold the 5 branches into one 3x3 kernel, packed for WMMA A-operand
// loads:  wpack[tap][icpair][oc] = { W_eff[oc][2*icpair][tap],
//                                    W_eff[oc][2*icpair+1][tap] }  (float2)
// ---------------------------------------------------------------------------
__global__ void fold_weights(const float* __restrict__ wsq,
                             const float* __restrict__ wver,   // (O,I,3,1)
                             const float* __restrict__ whor,   // (O,I,1,3)
                             const float* __restrict__ wd19,   // eye mask
                             const float* __restrict__ wd37,   // anti-diag mask
                             float* __restrict__ wpack) {
  int idx = blockIdx.x * blockDim.x + threadIdx.x;
  if (idx >= C_OUT * C_IN * 9) return;
  int tap = idx % 9;
  int ic  = (idx / 9) % C_IN;
  int oc  = idx / (9 * C_IN);
  int kh = tap / 3, kw = tap % 3;
  int oi = oc * C_IN + ic;

  float v = wsq[oi * 9 + tap];
  if (kw == 1)      v += wver[oi * 3 + kh];   // 3x1 branch -> center column
  if (kh == 1)      v += whor[oi * 3 + kw];   // 1x3 branch -> center row
  if (kh == kw)     v += wd19[oi * 9 + tap];  // main diagonal (eye)
  if (kw == 2 - kh) v += wd37[oi * 9 + tap];  // anti-diagonal

  wpack[(((tap * (C_IN / 2) + (ic >> 1)) * C_OUT) + oc) * 2 + (ic & 1)] = v;
}

// Issue async global->LDS copies for one 32-channel chunk (no VGPR data
// round-trip, tracked by ASYNCcnt). OOB halo elements are skipped; they
// keep the block-wide pre-zeroed value. Ordering vs the s_wait_asynccnt
// below is guaranteed by volatile-asm program order + __syncthreads fences.
__device__ __forceinline__ void stage_async(const float* __restrict__ x,
                                            int b, int icc, int x0, int y0,
                                            int tid, v2f* buf) {
  for (int i = tid; i < NPAIR * 18 * 18; i += 256) {
    int xx = i % 18;
    int yy = (i / 18) % 18;
    int p  = i / (18 * 18);
    int gy = y0 - 1 + yy;
    int gx = x0 - 1 + xx;
    if (gy >= 0 && gy < HH && gx >= 0 && gx < WW) {
      const float* g0 =
          &x[((size_t)(b * C_IN + icc + 2 * p) * HH + gy) * WW + gx];
      const float* g1 = g0 + (size_t)HH * WW;  // channel pair partner
      // LDS byte offset = low 32 bits of the flat shared-aperture address.
      unsigned l0 = (unsigned)(uintptr_t)&buf[(p * 18 + yy) * ROW_F2 + xx];
      unsigned l1 = l0 + 4;
      asm volatile("global_load_async_to_lds_b32 %0, %1, off"
                   :: "v"(l0), "v"((unsigned long long)(uintptr_t)g0));
      asm volatile("global_load_async_to_lds_b32 %0, %1, off"
                   :: "v"(l1), "v"((unsigned long long)(uintptr_t)g1));
    }
  }
}

// ---------------------------------------------------------------------------
// Phase 2: implicit-GEMM 3x3 conv.
//   block = 256 threads = 8 waves; block tile = 16x16 pixels of one image,
//   wave w owns output channels [16w, 16w+16). K-loop: 128 ic x 9 taps,
//   K=4 per WMMA. 16 rows x v8f accumulators persist across the 4 chunks.
// ---------------------------------------------------------------------------
__global__ __launch_bounds__(256) void conv3x3_wmma(
    const float* __restrict__ x,
    const float* __restrict__ wpack,
    float* __restrict__ out) {
  __shared__ v2f lds2[2 * BUF_F2];  // 110592 B (double buffered)

  const int tid  = threadIdx.x;
  const int lane = tid & 31;
  const int wave = tid >> 5;
  const int oc_base = wave * 16;

  const int bid = blockIdx.x;
  const int tx = bid % (WW / TILE);
  const int ty = (bid / (WW / TILE)) % (HH / TILE);
  const int b  = bid / ((WW / TILE) * (HH / TILE));
  const int x0 = tx * TILE;
  const int y0 = ty * TILE;

  const int n  = lane & 15;   // B-matrix N / A-matrix M index
  const int hi = lane >> 4;   // 0: K0..1 half, 1: K2..3 half

  v8f acc[16];
#pragma unroll
  for (int r = 0; r < 16; ++r) acc[r] = (v8f){0.f,0.f,0.f,0.f,0.f,0.f,0.f,0.f};

  const v2f* __restrict__ wp = (const v2f*)wpack;

  // Pre-zero both buffers once: async loads never touch OOB halo slots.
  {
    v2f z = (v2f){0.f, 0.f};
    for (int i = tid; i < 2 * BUF_F2; i += 256) lds2[i] = z;
  }
  __syncthreads();

  stage_async(x, b, 0, x0, y0, tid, &lds2[0]);  // prime chunk 0

  for (int c = 0; c < 4; ++c) {
    const int icc = c * ICCH;
    v2f* cur = &lds2[(c & 1) * BUF_F2];
    v2f* nxt = &lds2[((c + 1) & 1) * BUF_F2];

    // Chunk c resident: wait own async copies, then block-wide sync.
    asm volatile("s_wait_asynccnt 0x0" ::: "memory");
    __syncthreads();

    // Kick off chunk c+1 DMA; it overlaps the WMMA loop below.
    if (c < 3) stage_async(x, b, icc + ICCH, x0, y0, tid, nxt);

    const v2f* lane_base = cur + hi * PLANE_F2 + n;
    for (int tap = 0; tap < 9; ++tap) {
      const int dy = tap / 3, dx = tap % 3;

      // Hoist A operands for this tap (re-used across 16 output rows).
      v2f A[8];
#pragma unroll
      for (int q = 0; q < 8; ++q) {
        int icpair = ((icc + q * 4) >> 1) + hi;
        A[q] = wp[(tap * (C_IN / 2) + icpair) * C_OUT + oc_base + n];
      }

      const v2f* lt = lane_base + dy * ROW_F2 + dx;

      // 128 WMMAs (q = t>>4, r = t&15) with an explicit 8-deep rotating
      // B-prefetch pipeline: every ds_load_b64 has >= 8 WMMAs of slack.
#define BOFF(t) ((2 * ((t) >> 4)) * PLANE_F2 + ((t) & 15) * ROW_F2)
      v2f Bbuf[8];
#pragma unroll
      for (int t = 0; t < 8; ++t) Bbuf[t] = lt[BOFF(t)];
#pragma unroll
      for (int t = 0; t < 128; ++t) {
        v2f Bcur = Bbuf[t & 7];
        if (t + 8 < 128) Bbuf[t & 7] = lt[BOFF(t + 8)];
        // D = A(16x4) * B(4x16) + C  (full fp32)
        acc[t & 15] = __builtin_amdgcn_wmma_f32_16x16x4_f32(
            false, A[t >> 4], false, Bcur, (short)0, acc[t & 15], false, false);
      }
#undef BOFF
    }
  }

  // C/D layout: VGPR j -> M = j + 8*hi, N = lane&15.
#pragma unroll
  for (int r = 0; r < 16; ++r) {
#pragma unroll
    for (int j = 0; j < 8; ++j) {
      int oc = oc_base + hi * 8 + j;
      out[((size_t)(b * C_OUT + oc) * HH + (y0 + r)) * WW + x0 + n] = acc[r][j];
    }
  }
}

extern "C" void kernel_launch(void* const* d_in, const int* in_sizes, int n_in,
                              void* d_out, int out_size, void* d_ws, size_t ws_size,
                              hipStream_t stream) {
  const float* x    = (const float*)d_in[0];
  const float* wsq  = (const float*)d_in[1];
  const float* wver = (const float*)d_in[2];
  const float* whor = (const float*)d_in[3];
  const float* wd19 = (const float*)d_in[4];
  const float* wd37 = (const float*)d_in[5];
  float* out   = (float*)d_out;
  float* wpack = (float*)d_ws;  // 128*128*9*4 B = 576 KB, fully rewritten each call

  {
    int total = C_OUT * C_IN * 9;
    fold_weights<<<(total + 255) / 256, 256, 0, stream>>>(wsq, wver, whor,
                                                          wd19, wd37, wpack);
  }
  {
    int nblocks = BATCH * (HH / TILE) * (WW / TILE);  // 2304
    conv3x3_wmma<<<nblocks, 256, 0, stream>>>(x, wpack, out);
  }
}